// NGCFConv_69569880261269
// MI455X (gfx1250) — compile-verified
//
#include <hip/hip_runtime.h>
#include <hip/hip_bf16.h>
#include <math.h>
#include <stdint.h>

typedef float v2f __attribute__((ext_vector_type(2)));
typedef float v8f __attribute__((ext_vector_type(8)));

#define DIM 128
#define LDA 132                 // padded LDS row length (floats) to de-conflict banks
#define NEG_SLOPE 0.01f
#define NORM_EPS 1e-12f

// ---------------------------------------------------------------------------
// CDNA5 async global->LDS 16B copy (no VGPR round trip, tracked by ASYNCcnt).
// vdst operand = VGPR holding the workgroup-relative LDS byte address
// (generic shared-pointer low 32 bits, per flat-LDS truncation rule).
// ---------------------------------------------------------------------------
__device__ __forceinline__ void async_copy_b128_to_lds(void* lds_ptr, const void* gptr) {
    uint32_t lds_off = (uint32_t)(uintptr_t)lds_ptr;
    uint64_t gaddr   = (uint64_t)(uintptr_t)gptr;
    asm volatile("global_load_async_to_lds_b128 %0, %1, off"
                 :: "v"(lds_off), "v"(gaddr)
                 : "memory");
}

__device__ __forceinline__ void wait_asynccnt_zero() {
    asm volatile("s_wait_asynccnt 0x0" ::: "memory");
}

// ---------------------------------------------------------------------------
// CSR build: histogram -> exclusive scan -> scatter
// ---------------------------------------------------------------------------
__global__ void hist_kernel(const int* __restrict__ row, int* __restrict__ counts, int e) {
    int i = blockIdx.x * blockDim.x + threadIdx.x;
    if (i < e) atomicAdd(&counts[row[i]], 1);
}

// Single-block exclusive scan over n entries. 1024 threads, LDS Hillis-Steele.
__global__ void scan_exclusive_kernel(const int* __restrict__ counts,
                                      int* __restrict__ row_ptr, int n) {
    __shared__ int sm[1024];
    const int tid = threadIdx.x;
    int running = 0;
    for (int base = 0; base < n; base += 1024) {
        int i = base + tid;
        int v = (i < n) ? counts[i] : 0;
        sm[tid] = v;
        __syncthreads();
#pragma unroll
        for (int off = 1; off < 1024; off <<= 1) {
            int t = (tid >= off) ? sm[tid - off] : 0;
            __syncthreads();
            sm[tid] += t;
            __syncthreads();
        }
        int incl = sm[tid];
        if (i < n) row_ptr[i] = running + incl - v;   // exclusive prefix
        __syncthreads();
        running += sm[1023];
        __syncthreads();
    }
    if (tid == 0) row_ptr[n] = running;
}

__global__ void scatter_kernel(const int* __restrict__ row, const int* __restrict__ col,
                               const float* __restrict__ val, const int* __restrict__ row_ptr,
                               int* __restrict__ cursor, int* __restrict__ ccol,
                               float* __restrict__ cval, int e) {
    int i = blockIdx.x * blockDim.x + threadIdx.x;
    if (i >= e) return;
    int r = row[i];
    int pos = row_ptr[r] + atomicAdd(&cursor[r], 1);
    ccol[pos] = col[i];
    cval[pos] = val[i];
}

// ---------------------------------------------------------------------------
// Gather SpMM: one wave32 per output row, lane holds float4 (128 dims / 32 lanes).
// Per-edge 512B coalesced row gathers; embedding table is L2-resident (51MB << 192MB).
// ---------------------------------------------------------------------------
__global__ void spmm_kernel(const int* __restrict__ row_ptr, const int* __restrict__ ccol,
                            const float* __restrict__ cval, const float* __restrict__ ego,
                            float* __restrict__ side, int n) {
    int wave = (blockIdx.x * blockDim.x + threadIdx.x) >> 5;
    int lane = threadIdx.x & 31;
    if (wave >= n) return;
    int start = row_ptr[wave];
    int end   = row_ptr[wave + 1];
    float4 acc = make_float4(0.f, 0.f, 0.f, 0.f);
    for (int e = start; e < end; ++e) {
        int   c = ccol[e];           // uniform across wave -> scalar broadcast
        float v = cval[e];
        float4 x = ((const float4*)(ego + (size_t)c * DIM))[lane];
        acc.x += v * x.x; acc.y += v * x.y; acc.z += v * x.z; acc.w += v * x.w;
    }
    ((float4*)(side + (size_t)wave * DIM))[lane] = acc;
}

// ---------------------------------------------------------------------------
// BiAggregator: ego_out = leaky((ego+side)@Ws + bs) + leaky((ego*side)@Wp + bp)
// 256 threads = 8 waves; 128 rows per block; fp32 WMMA 16x16x4 chained over K=128.
// LDS: a_sum tile, a_prod tile, Ws, Wp  (4 * 128 * 132 floats = 270,336 B, dynamic).
// Weights are DMA'd via global_load_async_to_lds_b128 (ASYNCcnt), overlapped with
// VALU activation staging. Safe in-place: side rows staged to LDS before overwrite.
// ---------------------------------------------------------------------------
__global__ __launch_bounds__(256) void agg_wmma_kernel(
    const float* __restrict__ ego, const float* __restrict__ side,
    const float* __restrict__ w_sum, const float* __restrict__ b_sum,
    const float* __restrict__ w_prod, const float* __restrict__ b_prod,
    float* __restrict__ out, int n) {
    extern __shared__ float smem[];
    float* s_as = smem;                  // [128][LDA] ego+side
    float* s_ap = s_as + 128 * LDA;      // [128][LDA] ego*side
    float* s_ws = s_ap + 128 * LDA;      // [128][LDA] w_sum (K-major rows)
    float* s_wp = s_ws + 128 * LDA;      // [128][LDA] w_prod

    const int tid = threadIdx.x;
    const int r0  = blockIdx.x * 128;

    // Kick off async weight DMA: 4096 float4 per matrix, LDS-direct, no VGPRs.
    for (int idx = tid; idx < 4096; idx += 256) {
        int rr = idx >> 5;
        int c4 = (idx & 31) * 4;
        async_copy_b128_to_lds(s_ws + rr * LDA + c4, w_sum  + rr * DIM + c4);
        async_copy_b128_to_lds(s_wp + rr * LDA + c4, w_prod + rr * DIM + c4);
    }
    // Meanwhile stage activation tiles, computing sum/prod in flight (VALU path).
    for (int idx = tid; idx < 4096; idx += 256) {
        int rr = idx >> 5;
        int c4 = (idx & 31) * 4;
        int r  = r0 + rr;
        float4 e = make_float4(0.f, 0.f, 0.f, 0.f);
        float4 s = make_float4(0.f, 0.f, 0.f, 0.f);
        if (r < n) {
            e = *(const float4*)(ego  + (size_t)r * DIM + c4);
            s = *(const float4*)(side + (size_t)r * DIM + c4);
        }
        float4 a, m;
        a.x = e.x + s.x; a.y = e.y + s.y; a.z = e.z + s.z; a.w = e.w + s.w;
        m.x = e.x * s.x; m.y = e.y * s.y; m.z = e.z * s.z; m.w = e.w * s.w;
        *(float4*)(s_as + rr * LDA + c4) = a;
        *(float4*)(s_ap + rr * LDA + c4) = m;
    }
    wait_asynccnt_zero();
    __syncthreads();

    const int wave = tid >> 5;
    const int lane = tid & 31;
    const int m0   = wave * 16;
    // A fragment (16x4 f32): lanes 0-15 rows M, holding K={0,1}; lanes 16-31 K={2,3}.
    const int row_a = m0 + (lane & 15);
    const int kh    = (lane >> 4) * 2;
    // B fragment (4x16 f32): lanes 0-15 col N with K={0,1}; lanes 16-31 K={2,3}.
    const int ncol  = lane & 15;

    for (int nt = 0; nt < 8; ++nt) {
        const int n0 = nt * 16;
        v8f acc_s = {};
        v8f acc_p = {};
#pragma unroll
        for (int kt = 0; kt < 32; ++kt) {
            const int kb = kt * 4 + kh;
            v2f a_s = { s_as[row_a * LDA + kb], s_as[row_a * LDA + kb + 1] };
            v2f a_p = { s_ap[row_a * LDA + kb], s_ap[row_a * LDA + kb + 1] };
            v2f b_s = { s_ws[kb * LDA + n0 + ncol], s_ws[(kb + 1) * LDA + n0 + ncol] };
            v2f b_p = { s_wp[kb * LDA + n0 + ncol], s_wp[(kb + 1) * LDA + n0 + ncol] };
            acc_s = __builtin_amdgcn_wmma_f32_16x16x4_f32(
                false, a_s, false, b_s, (short)0, acc_s, false, false);
            acc_p = __builtin_amdgcn_wmma_f32_16x16x4_f32(
                false, a_p, false, b_p, (short)0, acc_p, false, false);
        }
        // C/D layout: lane holds N = n0+(lane&15); VGPR v holds M = v + 8*(lane>>4).
        const float bs = b_sum[n0 + ncol];
        const float bp = b_prod[n0 + ncol];
        const int mbase = r0 + m0 + (lane >> 4) * 8;
#pragma unroll
        for (int v = 0; v < 8; ++v) {
            int r = mbase + v;
            if (r < n) {
                float xs = acc_s[v] + bs;
                xs = (xs >= 0.f) ? xs : NEG_SLOPE * xs;
                float xp = acc_p[v] + bp;
                xp = (xp >= 0.f) ? xp : NEG_SLOPE * xp;
                out[(size_t)r * DIM + n0 + ncol] = xs + xp;
            }
        }
    }
}

// ---------------------------------------------------------------------------
// Row L2-normalize in place: one wave per row, float4 per lane, shuffle reduce.
// ---------------------------------------------------------------------------
__global__ void normalize_kernel(float* __restrict__ data, int n) {
    int row  = (blockIdx.x * blockDim.x + threadIdx.x) >> 5;
    int lane = threadIdx.x & 31;
    if (row >= n) return;
    float4 x = ((const float4*)(data + (size_t)row * DIM))[lane];
    float ss = x.x * x.x + x.y * x.y + x.z * x.z + x.w * x.w;
#pragma unroll
    for (int off = 16; off > 0; off >>= 1) ss += __shfl_xor(ss, off, 32);
    float nrm = sqrtf(ss);
    nrm = (nrm > NORM_EPS) ? nrm : NORM_EPS;
    float inv = 1.0f / nrm;
    x.x *= inv; x.y *= inv; x.z *= inv; x.w *= inv;
    ((float4*)(data + (size_t)row * DIM))[lane] = x;
}

// ---------------------------------------------------------------------------
static inline size_t align_up(size_t v, size_t a) { return (v + a - 1) & ~(a - 1); }

extern "C" void kernel_launch(void* const* d_in, const int* in_sizes, int n_in,
                              void* d_out, int out_size, void* d_ws, size_t ws_size,
                              hipStream_t stream) {
    const float* emb    = (const float*)d_in[0];
    const int*   arow   = (const int*)  d_in[1];
    const int*   acol   = (const int*)  d_in[2];
    const float* aval   = (const float*)d_in[3];
    const float* w_sum  = (const float*)d_in[4];
    const float* b_sum  = (const float*)d_in[5];
    const float* w_prod = (const float*)d_in[6];
    const float* b_prod = (const float*)d_in[7];

    const int N = in_sizes[0] / DIM;
    const int E = in_sizes[1];
    const int L = in_sizes[5] / DIM;      // 2

    float* out0 = (float*)d_out;
    float* out1 = out0 + (size_t)N * DIM;
    float* out2 = out1 + (size_t)N * DIM;

    // workspace carve (CSR only; activation scratch lives in the out slabs)
    char* ws = (char*)d_ws;
    int*   counts  = (int*)ws;                       ws += align_up((size_t)N * 4, 256);
    int*   cursor  = (int*)ws;                       ws += align_up((size_t)N * 4, 256);
    int*   row_ptr = (int*)ws;                       ws += align_up((size_t)(N + 1) * 4, 256);
    int*   ccol    = (int*)ws;                       ws += align_up((size_t)E * 4, 256);
    float* cval    = (float*)ws;                     ws += align_up((size_t)E * 4, 256);
    (void)ws_size; (void)n_in; (void)out_size; (void)L;

    // out[0] = input embeddings
    hipMemcpyAsync(out0, emb, (size_t)N * DIM * sizeof(float),
                   hipMemcpyDeviceToDevice, stream);

    // Build CSR on device (edges are constant, but launch must be stateless)
    hipMemsetAsync(counts, 0, (size_t)N * 4, stream);
    hipMemsetAsync(cursor, 0, (size_t)N * 4, stream);
    const int eb = (E + 255) / 256;
    hist_kernel<<<eb, 256, 0, stream>>>(arow, counts, E);
    scan_exclusive_kernel<<<1, 1024, 0, stream>>>(counts, row_ptr, N);
    scatter_kernel<<<eb, 256, 0, stream>>>(arow, acol, aval, row_ptr, cursor, ccol, cval, E);

    const int spmm_blocks = (N + 7) / 8;        // 8 waves/block, 1 row/wave
    const int agg_blocks  = (N + 127) / 128;
    const size_t agg_lds  = (size_t)4 * 128 * LDA * sizeof(float);  // 270,336 B

    // Layer 0: side1 -> out1, ego1 (un-normalized) in-place into out1
    spmm_kernel<<<spmm_blocks, 256, 0, stream>>>(row_ptr, ccol, cval, emb, out1, N);
    agg_wmma_kernel<<<agg_blocks, 256, agg_lds, stream>>>(
        emb, out1, w_sum, b_sum, w_prod, b_prod, out1, N);

    // Layer 1: side2 -> out2, ego2 (un-normalized) in-place into out2
    spmm_kernel<<<spmm_blocks, 256, 0, stream>>>(row_ptr, ccol, cval, out1, out2, N);
    agg_wmma_kernel<<<agg_blocks, 256, agg_lds, stream>>>(
        out1, out2, w_sum + DIM * DIM, b_sum + DIM, w_prod + DIM * DIM, b_prod + DIM,
        out2, N);

    // Now that un-normalized ego1 has been consumed, normalize both in place.
    normalize_kernel<<<spmm_blocks, 256, 0, stream>>>(out1, N);
    normalize_kernel<<<spmm_blocks, 256, 0, stream>>>(out2, N);
}